// Damping_57578331570470
// MI455X (gfx1250) — compile-verified
//
#include <hip/hip_runtime.h>
#include <math.h>

#define NQ     512
#define HIDD   1024
#define OUTQ   (NQ + NQ * (NQ - 1) / 2)   // 131328
#define SPLITK 4
#define KCHUNK (HIDD / SPLITK)            // 256

typedef __attribute__((ext_vector_type(2))) float v2f;
typedef __attribute__((ext_vector_type(4))) float v4f;
typedef __attribute__((ext_vector_type(8))) float v8f;

// ---------------------------------------------------------------------------
// y[j] = tanh(dot(x, W[:,j]) + b[j]);  W row-major (din x dout).
// Coalesced: lane j reads W[k*dout + j]. x staged in LDS (<= 4 KB).
// ---------------------------------------------------------------------------
__global__ void gemv_tanh_kernel(const float* __restrict__ x,
                                 const float* __restrict__ W,
                                 const float* __restrict__ b,
                                 float* __restrict__ y,
                                 int din, int dout) {
    __shared__ float xs[HIDD];
    for (int i = threadIdx.x; i < din; i += blockDim.x) xs[i] = x[i];
    __syncthreads();
    int j = blockIdx.x * blockDim.x + threadIdx.x;
    if (j < dout) {
        float s = b[j];
        for (int k = 0; k < din; ++k)
            s = fmaf(xs[k], W[(size_t)k * dout + j], s);
        y[j] = tanhf(s);
    }
}

// ---------------------------------------------------------------------------
// Split-K partial GEMV for o = h @ Wo.  Wo is 538 MB (>> 192 MB L2) ->
// stream with non-temporal 16B loads; 4 outputs/lane; k-chunk of 256 per
// block so ~4k waves are in flight (need ~14 MB outstanding to hit 23.3 TB/s;
// 8-deep unroll clauses 8 NT b128 loads per wave).
// Deterministic: partials reduced by a separate kernel, no atomics.
// ---------------------------------------------------------------------------
__global__ void gemv_out_partial_kernel(const float* __restrict__ h,
                                        const float* __restrict__ Wo,
                                        float* __restrict__ partial,
                                        int din, int dout) {
    __shared__ float hs[KCHUNK];
    const int kb = blockIdx.y;            // 0..SPLITK-1
    const int k0 = kb * KCHUNK;
    if (threadIdx.x < KCHUNK) hs[threadIdx.x] = h[k0 + threadIdx.x];
    __syncthreads();

    int j = (blockIdx.x * blockDim.x + threadIdx.x) * 4;
    if (j + 3 < dout) {
        v4f acc = {};
#pragma unroll 8
        for (int k = 0; k < KCHUNK; ++k) {
            v4f w = __builtin_nontemporal_load(
                        (const v4f*)(Wo + (size_t)(k0 + k) * dout + j));
            acc += hs[k] * w;             // contracts to v_fma
        }
        *(v4f*)(partial + (size_t)kb * dout + j) = acc;
    }
}

__global__ void reduce_out_kernel(const float* __restrict__ partial,
                                  const float* __restrict__ bo,
                                  float* __restrict__ o, int dout) {
    int j = blockIdx.x * blockDim.x + threadIdx.x;
    if (j < dout) {
        float s = bo[j];
#pragma unroll
        for (int p = 0; p < SPLITK; ++p) s += partial[(size_t)p * dout + j];
        o[j] = s;
    }
}

// ---------------------------------------------------------------------------
// L[r,c] = exp(o[r]) if r==c; o[N + r(r-1)/2 + c] if r>c; 0 otherwise.
// Writes the full 512x512 (ws is poisoned, so zeros must be explicit).
// ---------------------------------------------------------------------------
__global__ void build_L_kernel(const float* __restrict__ o,
                               float* __restrict__ L) {
    int idx = blockIdx.x * blockDim.x + threadIdx.x;
    if (idx >= NQ * NQ) return;
    int r = idx >> 9, c = idx & (NQ - 1);
    float v = 0.0f;
    if (r == c)      v = expf(o[r]);
    else if (r > c)  v = o[NQ + (r * (r - 1)) / 2 + c];
    L[idx] = v;
}

// ---------------------------------------------------------------------------
// D = L @ L^T via V_WMMA_F32_16X16X4_F32, one wave per 16x16 tile of D.
//
// A layout (16x4 f32, ISA 7.12.2): lane l holds row m0 + (l%16),
//   VGPR v holds k = k0 + 2*(l/16) + v  -> contiguous float2 at k0+2*half.
// B = L^T (4x16, K x N): B[k][n] = L[n0+n][k]  -> identical float2 load form
//   with row base n0.  C/D: VGPR i -> M = m0 + i + 8*(l/16), N = n0 + l%16.
//
// kmax = min(m0,n0)+16 (L[r][k]==0 for k>r) is a multiple of 16 -> manual
// x4 unroll is exact: 8 clause-batched b64 loads feed 4 back-to-back WMMAs.
// kmax is readfirstlane'd so the loop is SGPR-controlled and EXEC stays
// all-ones throughout (WMMA requirement).
// ---------------------------------------------------------------------------
__global__ void llt_wmma_kernel(const float* __restrict__ L,
                                float* __restrict__ D) {
    const int tileId = blockIdx.x * 4 + threadIdx.y;      // 0..1023
    const int m0 = (tileId >> 5) * 16;                    // tile row base
    const int n0 = (tileId & 31) * 16;                    // tile col base
    const int lane = threadIdx.x;                         // 0..31 (wave32)
    const int half = lane >> 4;                           // 0 or 1
    const int lm   = lane & 15;

    const float* __restrict__ pA = L + (size_t)(m0 + lm) * NQ + 2 * half;
    const float* __restrict__ pB = L + (size_t)(n0 + lm) * NQ + 2 * half;

    v8f acc = {};
    const int kmax =
        __builtin_amdgcn_readfirstlane((m0 < n0 ? m0 : n0) + 16);

    for (int k0 = 0; k0 < kmax; k0 += 16) {
        v2f a0 = *(const v2f*)(pA + k0);
        v2f a1 = *(const v2f*)(pA + k0 + 4);
        v2f a2 = *(const v2f*)(pA + k0 + 8);
        v2f a3 = *(const v2f*)(pA + k0 + 12);
        v2f b0 = *(const v2f*)(pB + k0);
        v2f b1 = *(const v2f*)(pB + k0 + 4);
        v2f b2 = *(const v2f*)(pB + k0 + 8);
        v2f b3 = *(const v2f*)(pB + k0 + 12);
        acc = __builtin_amdgcn_wmma_f32_16x16x4_f32(false, a0, false, b0,
                                                    (short)0, acc, false, false);
        acc = __builtin_amdgcn_wmma_f32_16x16x4_f32(false, a1, false, b1,
                                                    (short)0, acc, false, false);
        acc = __builtin_amdgcn_wmma_f32_16x16x4_f32(false, a2, false, b2,
                                                    (short)0, acc, false, false);
        acc = __builtin_amdgcn_wmma_f32_16x16x4_f32(false, a3, false, b3,
                                                    (short)0, acc, false, false);
    }

#pragma unroll
    for (int i = 0; i < 8; ++i) {
        int m = m0 + i + 8 * half;
        D[(size_t)m * NQ + n0 + lm] = acc[i];
    }
}

// ---------------------------------------------------------------------------
extern "C" void kernel_launch(void* const* d_in, const int* in_sizes, int n_in,
                              void* d_out, int out_size, void* d_ws, size_t ws_size,
                              hipStream_t stream) {
    const float* x  = (const float*)d_in[0];
    const float* W0 = (const float*)d_in[1];
    const float* b0 = (const float*)d_in[2];
    const float* W1 = (const float*)d_in[3];
    const float* b1 = (const float*)d_in[4];
    const float* W2 = (const float*)d_in[5];
    const float* b2 = (const float*)d_in[6];
    const float* Wo = (const float*)d_in[7];
    const float* bo = (const float*)d_in[8];
    float* D = (float*)d_out;

    float* ws = (float*)d_ws;
    float* h0      = ws;                       // 1024
    float* h1      = ws + HIDD;                // 1024
    float* h2      = ws + 2 * HIDD;            // 1024
    float* o       = ws + 3 * HIDD;            // 131328
    float* partial = ws + 3 * HIDD + OUTQ;     // SPLITK * 131328
    float* L       = partial;                  // 262144; overlays partial
                                               // (partial dead before L written)

    // 3-layer MLP (bandwidth-trivial)
    gemv_tanh_kernel<<<HIDD / 256, 256, 0, stream>>>(x,  W0, b0, h0, NQ,   HIDD);
    gemv_tanh_kernel<<<HIDD / 256, 256, 0, stream>>>(h0, W1, b1, h1, HIDD, HIDD);
    gemv_tanh_kernel<<<HIDD / 256, 256, 0, stream>>>(h1, W2, b2, h2, HIDD, HIDD);

    // Dominant 538 MB streaming GEMV: split-K x4 for occupancy, NT loads
    dim3 gridP((OUTQ / 4 + 255) / 256, SPLITK);      // 129 x 4 blocks
    gemv_out_partial_kernel<<<gridP, 256, 0, stream>>>(h2, Wo, partial,
                                                       HIDD, OUTQ);
    reduce_out_kernel<<<(OUTQ + 255) / 256, 256, 0, stream>>>(partial, bo, o,
                                                              OUTQ);

    // Scatter into lower-triangular L (zeros explicit; overwrites partials)
    build_L_kernel<<<(NQ * NQ) / 256, 256, 0, stream>>>(o, L);

    // D = L L^T with f32 WMMA, 32x32 tiles of 16x16, 4 waves/block
    dim3 blk(32, 4);
    llt_wmma_kernel<<<(32 * 32) / 4, blk, 0, stream>>>(L, D);
}